// ContrastiveLossNew_50697793962496
// MI455X (gfx1250) — compile-verified
//
#include <hip/hip_runtime.h>

// ContrastiveLoss: match L2 loss + non-match hinge, reduced to 3 scalars.
// Roofline: ~30 MB gathered traffic, ~10 MFLOP -> pure gather-bandwidth bound
// (~1.3 us at 23.3 TB/s). Strategy: one-pass b64/b128 gathered loads; the
// match-part D=16 squared-diff reduction runs on the WMMA unit as
// trace(X X^T) via 4 chained V_WMMA_F32_16X16X4_F32 (free FLOPs while waiting
// on memory, no cross-lane shuffles); VALU hinge for the non-match part;
// deterministic two-stage block reduction (no float atomics -> bit-stable
// across graph replays).

typedef __attribute__((ext_vector_type(2))) float v2f;
typedef __attribute__((ext_vector_type(8))) float v8f;

#define MARGIN       0.5f
#define NON_MATCH_W  1.0f
#define DIMS         16

// ---------------------------------------------------------------------------
// Match loss: each wave handles a tile of 16 pairs. diff tile X is 16x16;
// sum_p ||a_p - b_p||^2 == trace(X X^T), accumulated by 4 chained
// V_WMMA_F32_16X16X4_F32 ops (K=4 each). A-operand layout per CDNA5 ISA
// (32-bit A-matrix 16x4):
//   lane L  (<16): VGPR0 = X[L][4c+0], VGPR1 = X[L][4c+1]
//   lane L+16    : VGPR0 = X[L][4c+2], VGPR1 = X[L][4c+3]
// Passing the same registers for A and B yields the Gram matrix.
// ---------------------------------------------------------------------------
__global__ __launch_bounds__(256) void match_loss_wmma(
    const float* __restrict__ outA, const float* __restrict__ outB,
    const int* __restrict__ matchA, const int* __restrict__ matchB,
    int totalPairs, int N, int M, float* __restrict__ partials)
{
    __shared__ float red[256];
    const int tid  = threadIdx.x;
    const int lane = tid & 31;
    const int wave = tid >> 5;
    const int L    = lane & 15;
    const int half = lane >> 4;
    const int tiles = (totalPairs + 15) >> 4;
    const int tile  = blockIdx.x * 8 + wave;

    float t = 0.0f;
    if (tile < tiles) {                    // wave-uniform: EXEC all-ones inside
        const int p = tile * 16 + L;
        v2f a0 = {0.f, 0.f}, a1 = {0.f, 0.f}, a2 = {0.f, 0.f}, a3 = {0.f, 0.f};
        if (p < totalPairs) {              // lane-predicated loads only
            const int     b  = p / M;
            const size_t  ia = (size_t)matchA[p];
            const size_t  ib = (size_t)matchB[p];
            const float2* pa = reinterpret_cast<const float2*>(
                outA + (((size_t)b * (size_t)N + ia) << 4) + (size_t)(half << 1));
            const float2* pb = reinterpret_cast<const float2*>(
                outB + (((size_t)b * (size_t)N + ib) << 4) + (size_t)(half << 1));
            float2 xa, xb;
            xa = pa[0]; xb = pb[0]; a0 = (v2f){xa.x - xb.x, xa.y - xb.y};
            xa = pa[2]; xb = pb[2]; a1 = (v2f){xa.x - xb.x, xa.y - xb.y};
            xa = pa[4]; xb = pb[4]; a2 = (v2f){xa.x - xb.x, xa.y - xb.y};
            xa = pa[6]; xb = pb[6]; a3 = (v2f){xa.x - xb.x, xa.y - xb.y};
        }
        v8f c = {0.f, 0.f, 0.f, 0.f, 0.f, 0.f, 0.f, 0.f};
        // D = A*B + C with A == B == diff chunk -> Gram matrix accumulation
        c = __builtin_amdgcn_wmma_f32_16x16x4_f32(false, a0, false, a0, (short)0, c, false, false);
        c = __builtin_amdgcn_wmma_f32_16x16x4_f32(false, a1, false, a1, (short)0, c, false, false);
        c = __builtin_amdgcn_wmma_f32_16x16x4_f32(false, a2, false, a2, (short)0, c, false, false);
        c = __builtin_amdgcn_wmma_f32_16x16x4_f32(false, a3, false, a3, (short)0, c, false, false);
        // Trace extraction. Diag element (m,m): m<8 -> VGPR m, lane m;
        // m>=8 -> VGPR m-8, lane m+16 (i.e. lanes 24..31). So lane L holds a
        // diagonal element in c[L&7] iff L<8 or L>=24.
        const int  j      = lane & 7;
        float      cj     = 0.0f;
#pragma unroll
        for (int k = 0; k < 8; ++k) cj = (j == k) ? c[k] : cj;
        t = ((lane < 8) || (lane >= 24)) ? cj : 0.0f;
    }
    red[tid] = t;
    __syncthreads();
    for (int s = 128; s > 0; s >>= 1) {
        if (tid < s) red[tid] += red[tid + s];
        __syncthreads();
    }
    if (tid == 0) partials[blockIdx.x] = red[0];
}

// ---------------------------------------------------------------------------
// Non-match hinge: elementwise (not a matmul), one pair (2x16 floats) per
// thread via b128 loads.
// ---------------------------------------------------------------------------
__global__ __launch_bounds__(256) void nonmatch_loss(
    const float* __restrict__ outA, const float* __restrict__ outB,
    const int* __restrict__ nonA, const int* __restrict__ nonB,
    int totalPairs, int N, int Mn, float* __restrict__ partials)
{
    __shared__ float red[256];
    const int tid = threadIdx.x;
    const int g   = blockIdx.x * 256 + tid;
    float t = 0.0f;
    if (g < totalPairs) {
        const int     b  = g / Mn;
        const size_t  ia = (size_t)nonA[g];
        const size_t  ib = (size_t)nonB[g];
        const float4* pa = reinterpret_cast<const float4*>(outA + (((size_t)b * (size_t)N + ia) << 4));
        const float4* pb = reinterpret_cast<const float4*>(outB + (((size_t)b * (size_t)N + ib) << 4));
#pragma unroll
        for (int c = 0; c < 4; ++c) {
            const float4 va = pa[c];
            const float4 vb = pb[c];
            float d;
            d = va.x - vb.x; t += fmaxf(MARGIN - d * d, 0.0f);
            d = va.y - vb.y; t += fmaxf(MARGIN - d * d, 0.0f);
            d = va.z - vb.z; t += fmaxf(MARGIN - d * d, 0.0f);
            d = va.w - vb.w; t += fmaxf(MARGIN - d * d, 0.0f);
        }
    }
    red[tid] = t;
    __syncthreads();
    for (int s = 128; s > 0; s >>= 1) {
        if (tid < s) red[tid] += red[tid + s];
        __syncthreads();
    }
    if (tid == 0) partials[blockIdx.x] = red[0];
}

// ---------------------------------------------------------------------------
// Final deterministic reduction of block partials -> 3 output scalars.
// ---------------------------------------------------------------------------
__global__ void finalize_loss(const float* __restrict__ partials,
                              int nMatchBlocks, int nNonBlocks,
                              int M, int Mn, float* __restrict__ out)
{
    if (blockIdx.x == 0 && threadIdx.x == 0) {
        float ms = 0.0f, ns = 0.0f;
        for (int i = 0; i < nMatchBlocks; ++i) ms += partials[i];
        for (int i = 0; i < nNonBlocks; ++i)  ns += partials[nMatchBlocks + i];
        const float ml = ms / (float)M;
        const float nl = NON_MATCH_W * ns / (float)Mn;
        out[0] = ml + nl;   // contrastiveLossSum
        out[1] = ml;        // matchLossSum
        out[2] = nl;        // nonMatchLossSum
    }
}

extern "C" void kernel_launch(void* const* d_in, const int* in_sizes, int n_in,
                              void* d_out, int out_size, void* d_ws, size_t ws_size,
                              hipStream_t stream) {
    const float* outA   = (const float*)d_in[0];
    const float* outB   = (const float*)d_in[1];
    const int*   matchA = (const int*)d_in[2];
    const int*   matchB = (const int*)d_in[3];
    const int*   nonA   = (const int*)d_in[4];
    const int*   nonB   = (const int*)d_in[5];
    float*       out    = (float*)d_out;
    float*       wsf    = (float*)d_ws;

    const int B  = 4;                           // per reference setup
    const int N  = in_sizes[0] / (B * DIMS);    // 307200
    const int M  = in_sizes[2] / B;             // 5000
    const int Mn = in_sizes[4] / B;             // 50000
    const int totalMatch = in_sizes[2];         // B*M
    const int totalNon   = in_sizes[4];         // B*Mn

    const int tiles        = (totalMatch + 15) / 16;     // 16 pairs per wave tile
    const int matchBlocks  = (tiles + 7) / 8;            // 8 waves per block
    const int nonBlocks    = (totalNon + 255) / 256;

    match_loss_wmma<<<matchBlocks, 256, 0, stream>>>(
        outA, outB, matchA, matchB, totalMatch, N, M, wsf);

    nonmatch_loss<<<nonBlocks, 256, 0, stream>>>(
        outA, outB, nonA, nonB, totalNon, N, Mn, wsf + matchBlocks);

    finalize_loss<<<1, 32, 0, stream>>>(wsf, matchBlocks, nonBlocks, M, Mn, out);

    (void)in_sizes; (void)n_in; (void)out_size; (void)ws_size;
}